// QwenSparseMoeBlock_20813411516427
// MI455X (gfx1250) — compile-verified
//
#include <hip/hip_runtime.h>
#include <hip/hip_bf16.h>
#include <math.h>
#include <stdint.h>

typedef float v2f __attribute__((ext_vector_type(2)));
typedef float v8f __attribute__((ext_vector_type(8)));

#define T_TOKENS 8192
#define H_DIM    2048
#define E_EXP    16
#define TOPK     4
#define IMOE     1408
#define ISHARED  5632
#define SH_CHUNK 2048

// GEMM tiling: block = 256 threads = 8 waves, block tile 64x64,
// wave (wr 0..3, wc 0..1) owns 16 rows x 32 cols (two 16x16 accums per B-matrix).
#define TM 64
#define TN 64
#define KB 32            // K-slab per LDS stage
#define ASTRIDE 36       // 64x32 A tile, padded row stride (dwords)
#define PSTRIDE 132      // B pair-row stride: TN*2 + 4 (dwords)
#define PB (KB / 2)      // 16 pair-rows per B tile

__device__ __forceinline__ v8f wmma_f32(v2f a, v2f b, v8f c) {
  // D = A(16x4, f32) * B(4x16, f32) + C(16x16, f32)
  return __builtin_amdgcn_wmma_f32_16x16x4_f32(
      /*neg_a=*/false, a, /*neg_b=*/false, b,
      /*c_mod=*/(short)0, c, /*reuse_a=*/false, /*reuse_b=*/false);
}

// Async 16B copy global -> LDS (per-lane addresses; tracked by ASYNCcnt).
__device__ __forceinline__ void async_copy_b128(uint32_t lds_addr,
                                                const float* gptr) {
  asm volatile("global_load_async_to_lds_b128 %0, %1, off"
               :: "v"(lds_addr), "v"((unsigned long long)(uintptr_t)gptr)
               : "memory");
}
// Wait until at most N async copies outstanding for this wave.
__device__ __forceinline__ void wait_async_le2() {
  asm volatile("s_wait_asynccnt 0x2" ::: "memory");
}
__device__ __forceinline__ void wait_async_0() {
  asm volatile("s_wait_asynccnt 0x0" ::: "memory");
}

// ---------------------------------------------------------------------------
// Zero the output accumulator region [T,H] and the per-expert counters.
// ---------------------------------------------------------------------------
__global__ __launch_bounds__(256) void zero_kernel(float* __restrict__ out,
                                                   int* __restrict__ counts) {
  size_t i = ((size_t)blockIdx.x * blockDim.x + threadIdx.x) * 4;
  if (i < (size_t)T_TOKENS * H_DIM) {
    *(float4*)(out + i) = make_float4(0.f, 0.f, 0.f, 0.f);
  }
  if (blockIdx.x == 0 && threadIdx.x < E_EXP) counts[threadIdx.x] = 0;
}

// ---------------------------------------------------------------------------
// Router: one wave (32 lanes) per token. Computes logits (written to output),
// softmax + top-4 combine weights, and the sigmoid shared-expert gate.
// ---------------------------------------------------------------------------
__global__ __launch_bounds__(256) void router_kernel(
    const float* __restrict__ X, const float* __restrict__ RW,
    const float* __restrict__ SEG, float* __restrict__ logits,
    float* __restrict__ combine, float* __restrict__ sgate) {
  int lane = threadIdx.x & 31;
  int wave = threadIdx.x >> 5;
  int t = blockIdx.x * 8 + wave;
  if (t >= T_TOKENS) return;

  float acc[E_EXP];
#pragma unroll
  for (int e = 0; e < E_EXP; ++e) acc[e] = 0.f;
  float accs = 0.f;

  const float* xr = X + (size_t)t * H_DIM;
  for (int k = lane; k < H_DIM; k += 32) {
    float xv = xr[k];
    const float* rwk = RW + (size_t)k * E_EXP;
#pragma unroll
    for (int e = 0; e < E_EXP; ++e) acc[e] = fmaf(xv, rwk[e], acc[e]);
    accs = fmaf(xv, SEG[k], accs);
  }
#pragma unroll
  for (int e = 0; e < E_EXP; ++e) {
    for (int off = 16; off; off >>= 1) acc[e] += __shfl_xor(acc[e], off, 32);
  }
  for (int off = 16; off; off >>= 1) accs += __shfl_xor(accs, off, 32);

  if (lane == 0) {
    float mx = acc[0];
#pragma unroll
    for (int e = 1; e < E_EXP; ++e) mx = fmaxf(mx, acc[e]);
    float p[E_EXP];
    float sum = 0.f;
#pragma unroll
    for (int e = 0; e < E_EXP; ++e) { p[e] = __expf(acc[e] - mx); sum += p[e]; }
    float inv = 1.f / sum;
#pragma unroll
    for (int e = 0; e < E_EXP; ++e) p[e] *= inv;

    float w[E_EXP], q[E_EXP];
#pragma unroll
    for (int e = 0; e < E_EXP; ++e) { w[e] = 0.f; q[e] = p[e]; }
    for (int j = 0; j < TOPK; ++j) {   // ties -> lowest index, like lax.top_k
      int bi = 0; float bv = q[0];
      for (int e = 1; e < E_EXP; ++e) { if (q[e] > bv) { bv = q[e]; bi = e; } }
      w[bi] = p[bi]; q[bi] = -1.f;
    }
#pragma unroll
    for (int e = 0; e < E_EXP; ++e) {
      logits[(size_t)t * E_EXP + e]  = acc[e];
      combine[(size_t)t * E_EXP + e] = w[e];
    }
    sgate[t] = 1.f / (1.f + __expf(-accs));
  }
}

// ---------------------------------------------------------------------------
// Build per-expert token lists from dense combine weights.
// ---------------------------------------------------------------------------
__global__ __launch_bounds__(256) void gather_kernel(
    const float* __restrict__ combine, int* __restrict__ counts,
    int* __restrict__ lists) {
  int t = blockIdx.x * blockDim.x + threadIdx.x;
  if (t >= T_TOKENS) return;
  for (int e = 0; e < E_EXP; ++e) {
    if (combine[(size_t)t * E_EXP + e] > 0.f) {
      int pos = atomicAdd(&counts[e], 1);
      lists[(size_t)e * T_TOKENS + pos] = t;
    }
  }
}

// ---------------------------------------------------------------------------
// Stage 1: H1[m, i] = silu(Xg @ GW)[m,i] * (Xg @ UW)[m,i]
// A tile: double-buffered GLOBAL_LOAD_ASYNC_TO_LDS (per-lane gathered rows).
// B tile: register-staged, pair-interleaved so each WMMA B fragment is one
// aligned ds_load_b64.
// ---------------------------------------------------------------------------
__global__ __launch_bounds__(256) void stage1_kernel(
    const float* __restrict__ X, const float* __restrict__ GW,
    const float* __restrict__ UW, float* __restrict__ H1, int Icols,
    const int* __restrict__ gather, const int* __restrict__ cntPtr,
    int fixedCnt, int chunk0) {
  int cnt = cntPtr ? *cntPtr : fixedCnt;
  int m0 = blockIdx.x * TM;
  if (m0 >= cnt) return;
  int n0 = blockIdx.y * TN;

  __shared__ float Ab0[TM * ASTRIDE];
  __shared__ float Ab1[TM * ASTRIDE];
  __shared__ float Bg[PB * PSTRIDE];
  __shared__ float Bu[PB * PSTRIDE];

  int tid = threadIdx.x;
  int lane = tid & 31;
  int wave = tid >> 5;
  int wr = wave & 3, wc = wave >> 2;
  int half = lane >> 4, r = lane & 15;

  v8f cg0 = {}, cg1 = {}, cu0 = {}, cu1 = {};

  // A staging: thread -> (row 0..63, cols acol & acol+16), 2x b128 async.
  int arow = tid >> 2;
  int acol = (tid & 3) * 4;
  int am = m0 + arow;
  int asrc = (am < cnt) ? (gather ? gather[am] : (chunk0 + am))
                        : (gather ? gather[0] : chunk0);
  const float* Abase = X + (size_t)asrc * H_DIM;
  uint32_t ldsA0 = (uint32_t)(uintptr_t)&Ab0[arow * ASTRIDE + acol];
  uint32_t ldsA1 = (uint32_t)(uintptr_t)&Ab1[arow * ASTRIDE + acol];

  // B staging: thread -> pair-rows pr and pr+8 (k rows 2pr,2pr+1,2pr+16,2pr+17),
  // column pair (pc, pc+1).
  int pr = tid >> 5;
  int pc = (tid & 31) * 2;
  const float* G0 = GW + n0 + pc;
  const float* U0 = UW + n0 + pc;

  float2 g00, g01, g10, g11, u00, u01, u10, u11;

  // Prologue: async A slab 0 -> buffer 0; B slab 0 -> registers.
  async_copy_b128(ldsA0,      Abase + acol);
  async_copy_b128(ldsA0 + 64, Abase + acol + 16);
  {
    const float* g = G0 + (size_t)(2 * pr) * Icols;
    g00 = *(const float2*)(g);
    g01 = *(const float2*)(g + Icols);
    g10 = *(const float2*)(g + (size_t)16 * Icols);
    g11 = *(const float2*)(g + (size_t)17 * Icols);
    const float* u = U0 + (size_t)(2 * pr) * Icols;
    u00 = *(const float2*)(u);
    u01 = *(const float2*)(u + Icols);
    u10 = *(const float2*)(u + (size_t)16 * Icols);
    u11 = *(const float2*)(u + (size_t)17 * Icols);
  }

  for (int k0 = 0; k0 < H_DIM; k0 += KB) {
    int buf = (k0 >> 5) & 1;
    __syncthreads();  // previous slab's readers done (B LDS + other A buffer free)
    *(float4*)(Bg + pr * PSTRIDE + pc * 2)       = make_float4(g00.x, g01.x, g00.y, g01.y);
    *(float4*)(Bg + (pr + 8) * PSTRIDE + pc * 2) = make_float4(g10.x, g11.x, g10.y, g11.y);
    *(float4*)(Bu + pr * PSTRIDE + pc * 2)       = make_float4(u00.x, u01.x, u00.y, u01.y);
    *(float4*)(Bu + (pr + 8) * PSTRIDE + pc * 2) = make_float4(u10.x, u11.x, u10.y, u11.y);

    int kn = k0 + KB;
    bool more = kn < H_DIM;
    if (more) {  // issue async A for next slab into the other buffer
      uint32_t dst = buf ? ldsA0 : ldsA1;
      async_copy_b128(dst,      Abase + kn + acol);
      async_copy_b128(dst + 64, Abase + kn + acol + 16);
      wait_async_le2();  // current slab's 2 copies complete (in-order)
    } else {
      wait_async_0();
    }
    __syncthreads();  // B stores + all waves' async A visible

    if (more) {  // register-stage next B slab; overlaps with WMMA stream
      const float* g = G0 + (size_t)(kn + 2 * pr) * Icols;
      g00 = *(const float2*)(g);
      g01 = *(const float2*)(g + Icols);
      g10 = *(const float2*)(g + (size_t)16 * Icols);
      g11 = *(const float2*)(g + (size_t)17 * Icols);
      const float* u = U0 + (size_t)(kn + 2 * pr) * Icols;
      u00 = *(const float2*)(u);
      u01 = *(const float2*)(u + Icols);
      u10 = *(const float2*)(u + (size_t)16 * Icols);
      u11 = *(const float2*)(u + (size_t)17 * Icols);
    }

    int c0 = wc * 32;
    const float* Asel = buf ? Ab1 : Ab0;
    const float* arp = Asel + (wr * 16 + r) * ASTRIDE;
#pragma unroll
    for (int kk = 0; kk < KB; kk += 4) {
      int ka = kk + 2 * half;   // even
      int p  = ka >> 1;
      v2f a   = *(const v2f*)(arp + ka);
      v2f bg0 = *(const v2f*)(Bg + p * PSTRIDE + (c0 + r) * 2);
      v2f bg1 = *(const v2f*)(Bg + p * PSTRIDE + (c0 + 16 + r) * 2);
      v2f bu0 = *(const v2f*)(Bu + p * PSTRIDE + (c0 + r) * 2);
      v2f bu1 = *(const v2f*)(Bu + p * PSTRIDE + (c0 + 16 + r) * 2);
      cg0 = wmma_f32(a, bg0, cg0);
      cg1 = wmma_f32(a, bg1, cg1);
      cu0 = wmma_f32(a, bu0, cu0);
      cu1 = wmma_f32(a, bu1, cu1);
    }
  }

  // Epilogue: SwiGLU, write H1 (row-major [maxM, Icols]).
  int mbase = m0 + wr * 16 + 8 * half;  // C layout: VGPR v -> M = v + 8*half
  int cbase = n0 + wc * 32 + r;
#pragma unroll
  for (int v = 0; v < 8; ++v) {
    int m = mbase + v;
    float g0 = cg0[v], g1 = cg1[v];
    float h0 = cu0[v] * (g0 / (1.f + __expf(-g0)));
    float h1 = cu1[v] * (g1 / (1.f + __expf(-g1)));
    H1[(size_t)m * Icols + cbase]      = h0;
    H1[(size_t)m * Icols + cbase + 16] = h1;
  }
}

// ---------------------------------------------------------------------------
// Stage 2: OUT[tok, n] += scale[tok] * (H1 @ DW)[m, n]
// Stream-ordered kernels + unique writer per element => deterministic.
// ---------------------------------------------------------------------------
__global__ __launch_bounds__(256) void stage2_kernel(
    const float* __restrict__ H1, const float* __restrict__ DW,
    float* __restrict__ OUT, int Icols, const int* __restrict__ gather,
    const int* __restrict__ cntPtr, int fixedCnt, int chunk0,
    const float* __restrict__ scale, int sstride, int soff) {
  int cnt = cntPtr ? *cntPtr : fixedCnt;
  int m0 = blockIdx.x * TM;
  if (m0 >= cnt) return;
  int n0 = blockIdx.y * TN;

  __shared__ float Ab0[TM * ASTRIDE];
  __shared__ float Ab1[TM * ASTRIDE];
  __shared__ float Bs[PB * PSTRIDE];

  int tid = threadIdx.x;
  int lane = tid & 31;
  int wave = tid >> 5;
  int wr = wave & 3, wc = wave >> 2;
  int half = lane >> 4, r = lane & 15;

  v8f c0a = {}, c1a = {};

  int arow = tid >> 2;
  int acol = (tid & 3) * 4;
  int am = m0 + arow;  // H1 rows are local; padded rows hold finite junk
  const float* Abase = H1 + (size_t)am * Icols;
  uint32_t ldsA0 = (uint32_t)(uintptr_t)&Ab0[arow * ASTRIDE + acol];
  uint32_t ldsA1 = (uint32_t)(uintptr_t)&Ab1[arow * ASTRIDE + acol];

  int pr = tid >> 5;
  int pc = (tid & 31) * 2;
  const float* D0 = DW + n0 + pc;

  float2 d00, d01, d10, d11;

  async_copy_b128(ldsA0,      Abase + acol);
  async_copy_b128(ldsA0 + 64, Abase + acol + 16);
  {
    const float* d = D0 + (size_t)(2 * pr) * H_DIM;
    d00 = *(const float2*)(d);
    d01 = *(const float2*)(d + H_DIM);
    d10 = *(const float2*)(d + (size_t)16 * H_DIM);
    d11 = *(const float2*)(d + (size_t)17 * H_DIM);
  }

  for (int k0 = 0; k0 < Icols; k0 += KB) {
    int buf = (k0 >> 5) & 1;
    __syncthreads();
    *(float4*)(Bs + pr * PSTRIDE + pc * 2)       = make_float4(d00.x, d01.x, d00.y, d01.y);
    *(float4*)(Bs + (pr + 8) * PSTRIDE + pc * 2) = make_float4(d10.x, d11.x, d10.y, d11.y);

    int kn = k0 + KB;
    bool more = kn < Icols;
    if (more) {
      uint32_t dst = buf ? ldsA0 : ldsA1;
      async_copy_b128(dst,      Abase + kn + acol);
      async_copy_b128(dst + 64, Abase + kn + acol + 16);
      wait_async_le2();
    } else {
      wait_async_0();
    }
    __syncthreads();

    if (more) {
      const float* d = D0 + (size_t)(kn + 2 * pr) * H_DIM;
      d00 = *(const float2*)(d);
      d01 = *(const float2*)(d + H_DIM);
      d10 = *(const float2*)(d + (size_t)16 * H_DIM);
      d11 = *(const float2*)(d + (size_t)17 * H_DIM);
    }

    int c0 = wc * 32;
    const float* Asel = buf ? Ab1 : Ab0;
    const float* arp = Asel + (wr * 16 + r) * ASTRIDE;
#pragma unroll
    for (int kk = 0; kk < KB; kk += 4) {
      int ka = kk + 2 * half;
      int p  = ka >> 1;
      v2f a  = *(const v2f*)(arp + ka);
      v2f b0 = *(const v2f*)(Bs + p * PSTRIDE + (c0 + r) * 2);
      v2f b1 = *(const v2f*)(Bs + p * PSTRIDE + (c0 + 16 + r) * 2);
      c0a = wmma_f32(a, b0, c0a);
      c1a = wmma_f32(a, b1, c1a);
    }
  }

  int mbase = m0 + wr * 16 + 8 * half;
  int cbase = n0 + wc * 32 + r;
#pragma unroll
  for (int v = 0; v < 8; ++v) {
    int m = mbase + v;
    if (m < cnt) {
      int tok = gather ? gather[m] : (chunk0 + m);
      float s = scale[(size_t)tok * sstride + soff];
      size_t o = (size_t)tok * H_DIM + cbase;
      OUT[o]      += s * c0a[v];
      OUT[o + 16] += s * c1a[v];
    }
  }
}

// ---------------------------------------------------------------------------
// Host-side launcher.
// ---------------------------------------------------------------------------
extern "C" void kernel_launch(void* const* d_in, const int* in_sizes, int n_in,
                              void* d_out, int out_size, void* d_ws,
                              size_t ws_size, hipStream_t stream) {
  (void)in_sizes; (void)n_in; (void)out_size; (void)ws_size;

  const float* X   = (const float*)d_in[0];  // [T, H]
  const float* RW  = (const float*)d_in[1];  // [H, E]
  const float* EGW = (const float*)d_in[2];  // [E, H, IMOE]
  const float* EUW = (const float*)d_in[3];  // [E, H, IMOE]
  const float* EDW = (const float*)d_in[4];  // [E, IMOE, H]
  const float* SGW = (const float*)d_in[5];  // [H, ISHARED]
  const float* SUW = (const float*)d_in[6];  // [H, ISHARED]
  const float* SDW = (const float*)d_in[7];  // [ISHARED, H]
  const float* SEG = (const float*)d_in[8];  // [H, 1]

  float* out    = (float*)d_out;                   // [T, H] accumulator
  float* logits = out + (size_t)T_TOKENS * H_DIM;  // [T, E]

  char* ws = (char*)d_ws;
  float* combine = (float*)(ws + 0);        // T*E floats (512 KB)
  float* sgate   = (float*)(ws + 524288);   // T floats   (32 KB)
  int*   counts  = (int*)(ws + 557056);     // E ints
  int*   lists   = (int*)(ws + 557312);     // E*T ints   (512 KB)
  float* h1      = (float*)(ws + 1081600);  // 46.1 MB (T*IMOE == SH_CHUNK*ISHARED)

  zero_kernel<<<dim3((T_TOKENS * H_DIM) / (256 * 4)), dim3(256), 0, stream>>>(out, counts);
  router_kernel<<<dim3(T_TOKENS / 8), dim3(256), 0, stream>>>(X, RW, SEG, logits, combine, sgate);
  gather_kernel<<<dim3(T_TOKENS / 256), dim3(256), 0, stream>>>(combine, counts, lists);

  // Routed experts: grouped GEMMs, weights streamed once per expert.
  for (int e = 0; e < E_EXP; ++e) {
    const float* gw = EGW + (size_t)e * H_DIM * IMOE;
    const float* uw = EUW + (size_t)e * H_DIM * IMOE;
    const float* dw = EDW + (size_t)e * IMOE * H_DIM;
    const int* lst = lists + (size_t)e * T_TOKENS;
    stage1_kernel<<<dim3(T_TOKENS / TM, IMOE / TN), dim3(256), 0, stream>>>(
        X, gw, uw, h1, IMOE, lst, counts + e, 0, 0);
    stage2_kernel<<<dim3(T_TOKENS / TM, H_DIM / TN), dim3(256), 0, stream>>>(
        h1, dw, out, IMOE, lst, counts + e, 0, 0, combine, E_EXP, e);
  }

  // Shared expert in 4 token chunks (weights stay L2-resident across chunks).
  for (int c0 = 0; c0 < T_TOKENS; c0 += SH_CHUNK) {
    stage1_kernel<<<dim3(SH_CHUNK / TM, ISHARED / TN), dim3(256), 0, stream>>>(
        X, SGW, SUW, h1, ISHARED, nullptr, nullptr, SH_CHUNK, c0);
    stage2_kernel<<<dim3(SH_CHUNK / TM, H_DIM / TN), dim3(256), 0, stream>>>(
        h1, SDW, out, ISHARED, nullptr, nullptr, SH_CHUNK, c0, sgate, 1, 0);
  }
}